// NCLoss_34342558499545
// MI455X (gfx1250) — compile-verified
//
#include <hip/hip_runtime.h>

typedef __attribute__((ext_vector_type(2))) float v2f;
typedef __attribute__((ext_vector_type(4))) float v4f;
typedef __attribute__((ext_vector_type(8))) float v8f;

constexpr int N_ = 4, C_ = 3, K_ = 4, H_ = 128, W_ = 128;
constexpr int h_ = 64, w_ = 64, P_ = h_ * w_;           // 4096
constexpr int TILES_1D = P_ / 16;                        // 256
constexpr long TILES_TOTAL = (long)N_ * TILES_1D * TILES_1D; // 262144 waves

// ---------------- preprocessing: downsample + feature build -----------------
// images:  nearest, scale 0.5, half-pixel -> input pixel (2y+1, 2x+1)
// segs:    bilinear, scale 0.5, half-pixel -> exact 2x2 average pool
// feats[p] = [r,g,b]/0.15 ++ [y,x]/25, padded with zeros to 8 (K=4 WMMA steps)
__global__ void nc_prep(const float* __restrict__ img,
                        const float* __restrict__ seg,
                        const float* __restrict__ roi,
                        float* __restrict__ feats, float* __restrict__ sq,
                        float* __restrict__ Sv, float* __restrict__ s) {
  int t = blockIdx.x * blockDim.x + threadIdx.x;
  if (t >= N_ * P_) return;
  int n = t / P_, p = t % P_;
  int y = p / w_, x = p % w_;
  int iy = 2 * y + 1, ix = 2 * x + 1;

  const float inv_srgb = 1.0f / 0.15f;
  const float inv_sxy  = 1.0f / 25.0f;   // SIGMA_XY * SCALE_FACTOR = 25

  float f[8];
#pragma unroll
  for (int c = 0; c < C_; ++c)
    f[c] = img[((size_t)(n * C_ + c) * H_ + iy) * W_ + ix] * inv_srgb;
  f[3] = (float)y * inv_sxy;
  f[4] = (float)x * inv_sxy;
  f[5] = f[6] = f[7] = 0.0f;

  float sqv = 0.0f;
#pragma unroll
  for (int d = 0; d < 5; ++d) sqv = fmaf(f[d], f[d], sqv);

  float r = roi[((size_t)n * H_ + iy) * W_ + ix];

  float ssum = 0.0f;
#pragma unroll
  for (int k = 0; k < K_; ++k) {
    const float* sp = seg + ((size_t)(n * K_ + k) * H_ + 2 * y) * W_ + 2 * x;
    float v = 0.25f * (sp[0] + sp[1] + sp[W_] + sp[W_ + 1]) * r;
    Sv[((size_t)(n * P_ + p)) * 4 + k] = v;
    ssum += v;
  }
#pragma unroll
  for (int d = 0; d < 8; ++d)
    feats[((size_t)(n * P_ + p)) * 8 + d] = f[d];
  sq[n * P_ + p] = sqv;
  s[n * P_ + p]  = ssum;
}

// ---------------- main: one wave per 16x16 (p,q) tile -----------------------
__global__ void __launch_bounds__(256) nc_tiles(
    const float* __restrict__ feats, const float* __restrict__ sq,
    const float* __restrict__ Sv, const float* __restrict__ s,
    float* __restrict__ acc) {
  int lane = threadIdx.x & 31;
  int wib  = threadIdx.x >> 5;
  long waveId = (long)blockIdx.x * 8 + wib;

  int n   = (int)(waveId >> 16);          // 65536 tiles per batch image
  int rem = (int)(waveId & 65535);
  int p0  = (rem >> 8) << 4;
  int q0  = (rem & 255) << 4;

  int hi = lane >> 4;                     // lane half selects K-pair
  int li = lane & 15;                     // M (for A) / N (for B) index

  const float* fb  = feats + (size_t)n * P_ * 8;
  const float* svb = Sv    + (size_t)n * P_ * 4;
  const float* sqb = sq    + (size_t)n * P_;
  const float* sb  = s     + (size_t)n * P_;

  int pr = p0 + li, qr = q0 + li;

  // A (rows p) / B (cols q): lane holds K = {2*hi, 2*hi+1} (+kbase)
  v2f a0 = *(const v2f*)(fb + (size_t)pr * 8 + 2 * hi);
  v2f a1 = *(const v2f*)(fb + (size_t)pr * 8 + 4 + 2 * hi);
  v2f b0 = *(const v2f*)(fb + (size_t)qr * 8 + 2 * hi);
  v2f b1 = *(const v2f*)(fb + (size_t)qr * 8 + 4 + 2 * hi);

  v8f dots = {};
  dots = __builtin_amdgcn_wmma_f32_16x16x4_f32(false, a0, false, b0,
                                               (short)0, dots, false, false);
  dots = __builtin_amdgcn_wmma_f32_16x16x4_f32(false, a1, false, b1,
                                               (short)0, dots, false, false);

  // G = S_p . S_q^T  (K=4 -> single 16x16x4 step)
  v4f svp = *(const v4f*)(svb + (size_t)pr * 4);
  v4f svq = *(const v4f*)(svb + (size_t)qr * 4);
  v2f a2, b2;
  a2.x = hi ? svp.z : svp.x;  a2.y = hi ? svp.w : svp.y;
  b2.x = hi ? svq.z : svq.x;  b2.y = hi ? svq.w : svq.y;
  v8f g = {};
  g = __builtin_amdgcn_wmma_f32_16x16x4_f32(false, a2, false, b2,
                                            (short)0, g, false, false);

  // per-row/col norms; C/D layout: VGPR r, lane L -> M = r + 8*hi, N = li
  float sqq = sqb[q0 + li];
  v4f sqp0 = *(const v4f*)(sqb + p0 + 8 * hi);
  v4f sqp1 = *(const v4f*)(sqb + p0 + 8 * hi + 4);
  v4f sp0  = *(const v4f*)(sb  + p0 + 8 * hi);
  v4f sp1  = *(const v4f*)(sb  + p0 + 8 * hi + 4);
  float sqp[8] = {sqp0.x, sqp0.y, sqp0.z, sqp0.w, sqp1.x, sqp1.y, sqp1.z, sqp1.w};
  float spv[8] = {sp0.x,  sp0.y,  sp0.z,  sp0.w,  sp1.x,  sp1.y,  sp1.z,  sp1.w};

  float numA = 0.0f, denA = 0.0f;
#pragma unroll
  for (int r = 0; r < 8; ++r) {
    float wv = __expf(dots[r] - 0.5f * (sqp[r] + sqq));  // exp(-0.5||fp-fq||^2)
    numA = fmaf(wv, g[r], numA);
    denA = fmaf(wv, spv[r], denA);
  }

  // wave32 butterfly reduction
#pragma unroll
  for (int off = 16; off > 0; off >>= 1) {
    numA += __shfl_xor(numA, off, 32);
    denA += __shfl_xor(denA, off, 32);
  }

  __shared__ float sm[16];
  if (lane == 0) { sm[wib * 2] = numA; sm[wib * 2 + 1] = denA; }
  __syncthreads();
  if (threadIdx.x == 0) {
    float tn = 0.0f, td = 0.0f;
#pragma unroll
    for (int i = 0; i < 8; ++i) { tn += sm[i * 2]; td += sm[i * 2 + 1]; }
    atomicAdd(&acc[0], tn);
    atomicAdd(&acc[1], td);
  }
}

// ---------------- tiny helpers ----------------------------------------------
__global__ void nc_zero(float* acc) {
  if (threadIdx.x < 2) acc[threadIdx.x] = 0.0f;
}

__global__ void nc_final(const float* __restrict__ acc, float* __restrict__ out) {
  if (threadIdx.x == 0) {
    float num = acc[0];
    float den = acc[1] + 1e-7f;
    out[0] = -0.1f * (num / den) / (float)N_;  // WEIGHT * (-(num/den)/N)
  }
}

extern "C" void kernel_launch(void* const* d_in, const int* in_sizes, int n_in,
                              void* d_out, int out_size, void* d_ws, size_t ws_size,
                              hipStream_t stream) {
  const float* img = (const float*)d_in[0];  // [4,3,128,128]
  const float* seg = (const float*)d_in[1];  // [4,4,128,128]
  const float* roi = (const float*)d_in[2];  // [4,128,128]

  float* feats = (float*)d_ws;               // N*P*8  = 131072 floats
  float* sq    = feats + (size_t)N_ * P_ * 8; // N*P    =  16384
  float* Sv    = sq    + (size_t)N_ * P_;     // N*P*4  =  65536
  float* s     = Sv    + (size_t)N_ * P_ * 4; // N*P    =  16384
  float* acc   = s     + (size_t)N_ * P_;     // 2 floats

  nc_zero<<<1, 32, 0, stream>>>(acc);
  nc_prep<<<(N_ * P_ + 255) / 256, 256, 0, stream>>>(img, seg, roi,
                                                     feats, sq, Sv, s);
  nc_tiles<<<(int)(TILES_TOTAL / 8), 256, 0, stream>>>(feats, sq, Sv, s, acc);
  nc_final<<<1, 32, 0, stream>>>(acc, (float*)d_out);
}